// GAT_50337016709813
// MI455X (gfx1250) — compile-verified
//
#include <hip/hip_runtime.h>
#include <hip/hip_bf16.h>
#include <math.h>

typedef float v2f __attribute__((ext_vector_type(2)));
typedef float v8f __attribute__((ext_vector_type(8)));

#define NEG_SLOPE 0.2f

// ---------------------------------------------------------------------------
// Float atomic max via the signed/unsigned integer ordering trick.
// (init value must be -INF; works for mixed signs with concurrent updates)
// ---------------------------------------------------------------------------
__device__ __forceinline__ void atomicMaxF32(float* addr, float val) {
    if (val >= 0.0f) {
        atomicMax((int*)addr, __float_as_int(val));
    } else {
        atomicMin((unsigned int*)addr, __float_as_uint(val));
    }
}

// ---------------------------------------------------------------------------
// GEMM: Y[N,32] = X[N,K] @ W[K,32] (+ optional bias), f32 WMMA 16x16x4.
// One wave computes a 16-row x 32-col tile (two 16x16 accumulators).
// Block = 256 threads = 8 waves -> 128 rows per block.
// ---------------------------------------------------------------------------
__global__ void gat_gemm32_wmma(const float* __restrict__ X,
                                const float* __restrict__ W,
                                const float* __restrict__ bias,
                                float* __restrict__ Y,
                                int N, int K) {
    const int lane = threadIdx.x & 31;
    const int wave = threadIdx.x >> 5;
    const int m0 = (blockIdx.x * 8 + wave) * 16;
    if (m0 >= N) return;                 // wave-uniform: EXEC stays all-ones below

    v8f c0 = {};                         // cols 0..15
    v8f c1 = {};                         // cols 16..31

    // A-fragment addressing (16x4 f32): lanes 0-15 -> K=k0+0/1, lanes 16-31 -> K=k0+2/3
    const int mrow  = m0 + (lane & 15);
    const int mclmp = mrow < N ? mrow : (N - 1);
    const int khalf = (lane >> 4) * 2;
    const int nb    = lane & 15;

    for (int k0 = 0; k0 < K; k0 += 4) {
        const int ka = k0 + khalf;
        v2f a;
        a.x = X[(size_t)mclmp * K + ka];
        a.y = X[(size_t)mclmp * K + ka + 1];

        // B-fragment (4x16 f32): lanes 0-15 rows K=k0+0/1, lanes 16-31 rows K=k0+2/3
        v2f b0, b1v;
        b0.x  = W[(size_t)ka * 32 + nb];
        b0.y  = W[(size_t)(ka + 1) * 32 + nb];
        b1v.x = W[(size_t)ka * 32 + 16 + nb];
        b1v.y = W[(size_t)(ka + 1) * 32 + 16 + nb];

        c0 = __builtin_amdgcn_wmma_f32_16x16x4_f32(false, a, false, b0,  (short)0, c0, false, false);
        c1 = __builtin_amdgcn_wmma_f32_16x16x4_f32(false, a, false, b1v, (short)0, c1, false, false);
    }

    // C/D layout: VGPR v, lanes 0-15 -> row m0+v, lanes 16-31 -> row m0+8+v; col = lane&15
    const int rowoff = (lane >> 4) * 8;
    for (int v = 0; v < 8; ++v) {
        const int row = m0 + rowoff + v;
        if (row < N) {
            float y0 = c0[v];
            float y1 = c1[v];
            if (bias) { y0 += bias[nb]; y1 += bias[16 + nb]; }
            Y[(size_t)row * 32 + nb]      = y0;
            Y[(size_t)row * 32 + 16 + nb] = y1;
        }
    }
}

// ---------------------------------------------------------------------------
// Per-node attention logits: as[n] = h[n,:].a_src, ad[n] = h[n,:].a_dst
// One wave per node, lane = feature, cross-lane reduce.
// ---------------------------------------------------------------------------
__global__ void gat_alpha(const float* __restrict__ h,
                          const float* __restrict__ a_src,
                          const float* __restrict__ a_dst,
                          float* __restrict__ out_s,
                          float* __restrict__ out_d,
                          int N) {
    const int lane = threadIdx.x & 31;
    const int node = (int)(((size_t)blockIdx.x * blockDim.x + threadIdx.x) >> 5);
    if (node >= N) return;
    const float hv = h[(size_t)node * 32 + lane];
    float s = hv * a_src[lane];
    float d = hv * a_dst[lane];
    for (int off = 16; off; off >>= 1) {
        s += __shfl_xor(s, off, 32);
        d += __shfl_xor(d, off, 32);
    }
    if (lane == 0) { out_s[node] = s; out_d[node] = d; }
}

// ---------------------------------------------------------------------------
// Init: agg = 0 (N*32), mmax = -inf, denom = 0 (N)
// ---------------------------------------------------------------------------
__global__ void gat_init(float* __restrict__ agg,
                         float* __restrict__ mmax,
                         float* __restrict__ denom,
                         int N) {
    const size_t tid = (size_t)blockIdx.x * blockDim.x + threadIdx.x;
    if (tid < (size_t)N * 32) agg[tid] = 0.0f;
    if (tid < (size_t)N) { mmax[tid] = -INFINITY; denom[tid] = 0.0f; }
}

// ---------------------------------------------------------------------------
// Edge pass 1: e = leaky_relu(as[src] + ad[dst]); segment max over dst.
// Edges [0,E) from edge_index, [E, E+N) are self loops.
// ---------------------------------------------------------------------------
__global__ void gat_edge_max(const int* __restrict__ ei,
                             const float* __restrict__ as_,
                             const float* __restrict__ ad_,
                             float* __restrict__ ebuf,
                             float* __restrict__ mmax,
                             long E, long T) {
    const long tid = (long)blockIdx.x * blockDim.x + threadIdx.x;
    if (tid >= T) return;
    int s, d;
    if (tid < E) { s = ei[tid]; d = ei[E + tid]; }
    else         { s = d = (int)(tid - E); }
    float ev = as_[s] + ad_[d];
    ev = ev > 0.0f ? ev : NEG_SLOPE * ev;
    ebuf[tid] = ev;
    atomicMaxF32(&mmax[d], ev);
}

// ---------------------------------------------------------------------------
// Edge pass 2: ex = exp(e - mmax[dst]); segment sum into denom.
// ---------------------------------------------------------------------------
__global__ void gat_edge_exp(const int* __restrict__ ei,
                             const float* __restrict__ mmax,
                             float* __restrict__ ebuf,
                             float* __restrict__ denom,
                             long E, long T) {
    const long tid = (long)blockIdx.x * blockDim.x + threadIdx.x;
    if (tid >= T) return;
    const int d = (tid < E) ? ei[E + tid] : (int)(tid - E);
    const float ex = __expf(ebuf[tid] - mmax[d]);
    ebuf[tid] = ex;
    atomicAdd(&denom[d], ex);
}

// ---------------------------------------------------------------------------
// Edge pass 3: agg[dst,:] += (ex/denom[dst]) * h[src,:].
// One wave per edge; lane = feature -> coalesced 128B gather + 128B atomic.
// ---------------------------------------------------------------------------
__global__ void gat_edge_aggregate(const int* __restrict__ ei,
                                   const float* __restrict__ ebuf,
                                   const float* __restrict__ denom,
                                   const float* __restrict__ h,
                                   float* __restrict__ agg,
                                   long E, long T) {
    const int lane = threadIdx.x & 31;
    const long eidx = (long)(((size_t)blockIdx.x * blockDim.x + threadIdx.x) >> 5);
    if (eidx >= T) return;
    int s, d;
    if (eidx < E) { s = ei[eidx]; d = ei[E + eidx]; }
    else          { s = d = (int)(eidx - E); }
    const float alpha = ebuf[eidx] / denom[d];
    const float val = alpha * h[(size_t)s * 32 + lane];
    atomicAdd(&agg[(size_t)d * 32 + lane], val);
}

// ---------------------------------------------------------------------------
// Finalize layer: z = relu(agg + bias), in place.
// ---------------------------------------------------------------------------
__global__ void gat_bias_relu(float* __restrict__ agg,
                              const float* __restrict__ bias,
                              int N) {
    const size_t tid = (size_t)blockIdx.x * blockDim.x + threadIdx.x;
    if (tid < (size_t)N * 32) {
        const float v = agg[tid] + bias[tid & 31];
        agg[tid] = v > 0.0f ? v : 0.0f;
    }
}

// ---------------------------------------------------------------------------
extern "C" void kernel_launch(void* const* d_in, const int* in_sizes, int n_in,
                              void* d_out, int out_size, void* d_ws, size_t ws_size,
                              hipStream_t stream) {
    const float* x   = (const float*)d_in[0];
    const int*   ei  = (const int*)  d_in[1];
    const float* W1  = (const float*)d_in[2];
    const float* a1s = (const float*)d_in[3];
    const float* a1d = (const float*)d_in[4];
    const float* b1  = (const float*)d_in[5];
    const float* W2  = (const float*)d_in[6];
    const float* a2s = (const float*)d_in[7];
    const float* a2d = (const float*)d_in[8];
    const float* b2  = (const float*)d_in[9];
    const float* Wf  = (const float*)d_in[10];
    const float* bf  = (const float*)d_in[11];

    const int  H   = in_sizes[3];           // 32
    const int  Fin = in_sizes[2] / H;       // 128
    const int  N   = in_sizes[0] / Fin;     // 100000
    const long E   = (long)in_sizes[1] / 2; // 3.2M
    const long T   = E + N;                 // edges + self loops

    // Workspace layout (floats): ~41 MB total
    float* ws    = (float*)d_ws;
    float* bufA  = ws;                         // h  [N,32]
    float* bufB  = bufA + (size_t)N * 32;      // agg / z [N,32]
    float* asv   = bufB + (size_t)N * 32;      // [N]
    float* adv   = asv + N;                    // [N]
    float* mmax  = adv + N;                    // [N]
    float* denom = mmax + N;                   // [N]
    float* ebuf  = denom + N;                  // [E+N]

    const dim3 blk(256);
    const int gGemm     = (N + 127) / 128;
    const int gNode32   = (int)(((size_t)N * 32 + 255) / 256);
    const int gWaveNode = (N + 7) / 8;
    const int gEdge     = (int)((T + 255) / 256);
    const int gWaveEdge = (int)((T + 7) / 8);

    // ---------------- Layer 1 ----------------
    gat_gemm32_wmma<<<gGemm, blk, 0, stream>>>(x, W1, nullptr, bufA, N, Fin);
    gat_alpha<<<gWaveNode, blk, 0, stream>>>(bufA, a1s, a1d, asv, adv, N);
    gat_init<<<gNode32, blk, 0, stream>>>(bufB, mmax, denom, N);
    gat_edge_max<<<gEdge, blk, 0, stream>>>(ei, asv, adv, ebuf, mmax, E, T);
    gat_edge_exp<<<gEdge, blk, 0, stream>>>(ei, mmax, ebuf, denom, E, T);
    gat_edge_aggregate<<<gWaveEdge, blk, 0, stream>>>(ei, ebuf, denom, bufA, bufB, E, T);
    gat_bias_relu<<<gNode32, blk, 0, stream>>>(bufB, b1, N);      // bufB = z1

    // ---------------- Layer 2 ----------------
    gat_gemm32_wmma<<<gGemm, blk, 0, stream>>>(bufB, W2, nullptr, bufA, N, H);
    gat_alpha<<<gWaveNode, blk, 0, stream>>>(bufA, a2s, a2d, asv, adv, N);
    gat_init<<<gNode32, blk, 0, stream>>>(bufB, mmax, denom, N);  // z1 consumed; reuse as agg
    gat_edge_max<<<gEdge, blk, 0, stream>>>(ei, asv, adv, ebuf, mmax, E, T);
    gat_edge_exp<<<gEdge, blk, 0, stream>>>(ei, mmax, ebuf, denom, E, T);
    gat_edge_aggregate<<<gWaveEdge, blk, 0, stream>>>(ei, ebuf, denom, bufA, bufB, E, T);
    gat_bias_relu<<<gNode32, blk, 0, stream>>>(bufB, b2, N);      // bufB = z2

    // ---------------- Final linear: out = z2 @ Wf + bf ----------------
    gat_gemm32_wmma<<<gGemm, blk, 0, stream>>>(bufB, Wf, bf, (float*)d_out, N, H);
}